// LTVStateSpace_66743791780174
// MI455X (gfx1250) — compile-verified
//
#include <hip/hip_runtime.h>
#include <math.h>
#include <stdint.h>

// MI455X / gfx1250: wave32, f32 WMMA 16x16x4, 320KB LDS per WGP,
// GLOBAL_LOAD_ASYNC_TO_LDS (ASYNCcnt) for direct memory->LDS staging.
// Persistent single-workgroup Kalman filter (information-filter form),
// double-buffered async L_t tile.

typedef __attribute__((ext_vector_type(2))) float v2f;
typedef __attribute__((ext_vector_type(8))) float v8f;

#define TSTEPS   2048
#define NOBS     256
#define KDIM     64
#define LDA      68          // padded stride for 64x64 LDS matrices
#define LDL      68          // padded stride for 256x64 L tile
#define NTHREADS 512         // 16 waves: one 16x16 WMMA tile per wave

// ---------------------------------------------------------------------------
// 16x16 output tile per wave, D = A(64xKin) * B(Kinx64) via V_WMMA_F32_16X16X4_F32.
// ---------------------------------------------------------------------------
template <typename LA, typename LB>
__device__ inline void wmma_gemm64(float* __restrict__ out, int ldo, int inner,
                                   LA la, LB lb) {
  const int lane = threadIdx.x & 31;
  const int wid  = threadIdx.x >> 5;
  if (wid >= 16) return;                 // never taken at blockDim=512
  const int tm   = (wid >> 2) << 4;
  const int tn   = (wid & 3) << 4;
  const int half = lane >> 4;
  const int lr   = lane & 15;
  v8f acc = {0.f, 0.f, 0.f, 0.f, 0.f, 0.f, 0.f, 0.f};
  for (int kk = 0; kk < inner; kk += 4) {
    const int ka = kk + 2 * half;
    v2f a, b;
    a.x = la(tm + lr, ka);
    a.y = la(tm + lr, ka + 1);
    b.x = lb(ka, tn + lr);
    b.y = lb(ka + 1, tn + lr);
    acc = __builtin_amdgcn_wmma_f32_16x16x4_f32(
        /*neg_a=*/false, a, /*neg_b=*/false, b,
        /*c_mod=*/(short)0, acc, /*reuse_a=*/false, /*reuse_b=*/false);
  }
#pragma unroll
  for (int r = 0; r < 8; ++r)
    out[(tm + r + 8 * half) * ldo + tn + lr] = acc[r];
}

// ---------------------------------------------------------------------------
// Direct memory->LDS async fill of one 256x64 L tile (16B granules, per-lane).
// 4096 B128 transfers total: 512 lanes x 8 issues. Tracked by ASYNCcnt.
// ---------------------------------------------------------------------------
__device__ inline void issue_L_async(uint32_t lds_base_bytes,
                                     const float* __restrict__ Lg) {
  const int tid = threadIdx.x;
#pragma unroll
  for (int k = 0; k < (NOBS * KDIM / 4) / NTHREADS; ++k) {   // 8
    const int c    = tid + k * NTHREADS;
    const int row  = c >> 4;           // 16 x 16B chunks per 64-float row
    const int col4 = c & 15;
    const uint32_t lofs = lds_base_bytes + (uint32_t)(row * LDL + col4 * 4) * 4u;
    const float* g = Lg + row * KDIM + col4 * 4;
    asm volatile("global_load_async_to_lds_b128 %0, %1, off"
                 :: "v"(lofs), "v"(g) : "memory");
  }
}

__device__ inline void wait_async0() {
  asm volatile("s_wait_asynccnt 0" ::: "memory");
}

// ---------------------------------------------------------------------------
// In-place right-looking Cholesky of 64x64 SPD matrix in LDS, wave 0 only.
// Lockstep wave32 + in-order per-wave LDS ops; wave_barrier fences compiler.
// ---------------------------------------------------------------------------
__device__ inline void chol64_wave0(float* __restrict__ M, float* __restrict__ logdet) {
  if (threadIdx.x >= 32) return;
  const int lane = threadIdx.x;
  float ld2 = 0.f;
  for (int j = 0; j < KDIM; ++j) {
    const float d = sqrtf(fmaxf(M[j * LDA + j], 1e-20f));
    ld2 += 2.f * logf(d);
    if (lane == 0) M[j * LDA + j] = d;
    const float inv = 1.f / d;
    for (int i = j + 1 + lane; i < KDIM; i += 32) M[i * LDA + j] *= inv;
    __builtin_amdgcn_wave_barrier();
    const int n = KDIM - 1 - j;
    for (int x = lane; x < n * n; x += 32) {
      const int i = j + 1 + x / n;
      const int p = j + 1 + x % n;
      M[i * LDA + p] -= M[i * LDA + j] * M[p * LDA + j];
    }
    __builtin_amdgcn_wave_barrier();
  }
  if (lane == 0) *logdet = ld2;
}

// Lower-triangular inverse X = C^{-1}; one column per lane (2 cols each).
__device__ inline void trinv64_wave0(const float* __restrict__ C, float* __restrict__ X) {
  if (threadIdx.x >= 32) return;
  const int lane = threadIdx.x;
  for (int s = 0; s < 2; ++s) {
    const int c = lane + 32 * s;
    for (int i = 0; i < c; ++i) X[i * LDA + c] = 0.f;
    for (int i = c; i < KDIM; ++i) {
      float sum = (i == c) ? 1.f : 0.f;
      for (int p = c; p < i; ++p) sum -= C[i * LDA + p] * X[p * LDA + c];
      X[i * LDA + c] = sum / C[i * LDA + i];
    }
  }
}

__global__ __launch_bounds__(NTHREADS, 1)
void kalman_persistent(const float* __restrict__ obs,     // [T,N]
                       const float* __restrict__ Lam,     // [T,N,K]
                       const float* __restrict__ A_raw,   // [K,K]
                       const float* __restrict__ lQ,      // [K]
                       const float* __restrict__ lR,      // [N]
                       float* __restrict__ out) {         // [T,K+1]
  extern __shared__ float sm[];
  float* A_s   = sm;                       // 64x68  scaled transition
  float* P_s   = A_s   + KDIM * LDA;       // 64x68  posterior covariance
  float* Pp    = P_s   + KDIM * LDA;       // 64x68  P_pred -> chol factor
  float* T1    = Pp    + KDIM * LDA;       // 64x68  temp / P_pred^{-1}
  float* Ci    = T1    + KDIM * LDA;       // 64x68  triangular inverse
  float* G     = Ci    + KDIM * LDA;       // 64x68  L^T R^{-1} L -> S -> chol(S)
  float* Lbuf  = G     + KDIM * LDA;       // 2 x 256x68 double-buffered L tile
  float* rinvR = Lbuf  + 2 * NOBS * LDL;   // 256
  float* rinv  = rinvR + NOBS;             // 256
  float* v     = rinv  + NOBS;             // 256
  float* u     = v     + NOBS;             // 64
  float* w1    = u     + KDIM;             // 64
  float* wv    = w1    + KDIM;             // 64
  float* z     = wv    + KDIM;             // 64
  float* zp    = z     + KDIM;             // 64
  float* qd    = zp    + KDIM;             // 64
  float* xv    = qd    + KDIM;             // 64
  float* red   = xv    + KDIM;             // 16

  const int tid = threadIdx.x;
  const float LOG2PI = 1.8378770664093453f;
  const uint32_t lbuf_b = (uint32_t)(uintptr_t)Lbuf;   // LDS byte offset (addr[31:0])

  // ------------------------- prologue (once) ------------------------------
  issue_L_async(lbuf_b, Lam);              // start L[0] fill; overlaps prologue

  for (int i = tid; i < KDIM * KDIM; i += NTHREADS)
    A_s[(i >> 6) * LDA + (i & 63)] = A_raw[i];
  for (int i = tid; i < KDIM * KDIM; i += NTHREADS)
    P_s[(i >> 6) * LDA + (i & 63)] = ((i >> 6) == (i & 63)) ? 1.f : 0.f;
  if (tid < KDIM) { qd[tid] = expf(lQ[tid]); z[tid] = 0.f; xv[tid] = 1.f; }
  if (tid < NOBS) rinvR[tid] = 1.f / (expf(lR[tid]) + 1e-4f);
  if (tid == 0) {
    float s = 0.f;
    for (int n = 0; n < NOBS; ++n) s += logf(expf(lR[n]) + 1e-4f);
    red[3] = s;                                       // sum log diag(R')
  }
  __syncthreads();

  // power iteration: sigma_max(A_raw) via x <- normalize(A^T A x)
  for (int it = 0; it < 32; ++it) {
    if (tid < KDIM) {
      float s = 0.f;
      for (int j = 0; j < KDIM; ++j) s += A_s[tid * LDA + j] * xv[j];
      w1[tid] = s;
    }
    __syncthreads();
    if (tid < KDIM) {
      float s = 0.f;
      for (int j = 0; j < KDIM; ++j) s += A_s[j * LDA + tid] * w1[j];
      wv[tid] = s;
    }
    __syncthreads();
    if (tid == 0) {
      float s = 0.f;
      for (int j = 0; j < KDIM; ++j) s += wv[j] * wv[j];
      red[0] = 1.f / (sqrtf(s) + 1e-30f);
    }
    __syncthreads();
    if (tid < KDIM) xv[tid] = wv[tid] * red[0];
    __syncthreads();
  }
  if (tid < KDIM) {
    float s = 0.f;
    for (int j = 0; j < KDIM; ++j) s += A_s[tid * LDA + j] * xv[j];
    w1[tid] = s;
  }
  __syncthreads();
  if (tid == 0) {
    float s = 0.f;
    for (int j = 0; j < KDIM; ++j) s += w1[j] * w1[j];
    red[0] = 0.98f / (sqrtf(s) + 1e-6f);              // MAX_SR / (sigma+1e-6)
  }
  __syncthreads();
  for (int i = tid; i < KDIM * KDIM; i += NTHREADS)
    A_s[(i >> 6) * LDA + (i & 63)] *= red[0];
  __syncthreads();

  // ------------------------- sequential scan ------------------------------
  for (int t = 0; t < TSTEPS; ++t) {
    // wait for this step's async L tile; previous step's barriers already
    // guarantee the other buffer is no longer being read.
    wait_async0();
    __syncthreads();
    if (t + 1 < TSTEPS)                                // overlap next fill
      issue_L_async(lbuf_b + (uint32_t)(((t + 1) & 1) * NOBS * LDL) * 4u,
                    Lam + (size_t)(t + 1) * (NOBS * KDIM));
    const float* L_s = Lbuf + (t & 1) * (NOBS * LDL);

    if (tid < KDIM) {                                  // z_pred = A z
      float s = 0.f;
      for (int j = 0; j < KDIM; ++j) s += A_s[tid * LDA + j] * z[j];
      zp[tid] = s;
    }
    __syncthreads();

    // T1 = A @ P
    wmma_gemm64(T1, LDA, KDIM,
        [&](int m, int p) { return A_s[m * LDA + p]; },
        [&](int p, int n) { return P_s[p * LDA + n]; });
    __syncthreads();
    // Pp = T1 @ A^T  (+ diag Q after)
    wmma_gemm64(Pp, LDA, KDIM,
        [&](int m, int p) { return T1[m * LDA + p]; },
        [&](int p, int n) { return A_s[n * LDA + p]; });
    __syncthreads();
    if (tid < KDIM) Pp[tid * LDA + tid] += qd[tid];
    if (tid < NOBS) {                                  // innovation + mask
      const float y = obs[(size_t)t * NOBS + tid];
      const bool miss = (y != y);                      // NaN check
      float s = 0.f;
      for (int j = 0; j < KDIM; ++j) s += L_s[tid * LDL + j] * zp[j];
      v[tid]    = miss ? 0.f : (y - s);
      rinv[tid] = miss ? 0.f : rinvR[tid];
    }
    __syncthreads();

    // G = L^T diag(rinv) L   (64x64, inner 256, WMMA)
    wmma_gemm64(G, LDA, NOBS,
        [&](int m, int p) { return L_s[p * LDL + m]; },
        [&](int p, int n) { return L_s[p * LDL + n] * rinv[p]; });
    if (tid < KDIM) {                                  // u = L^T R^{-1} v
      float s = 0.f;
      for (int n = 0; n < NOBS; ++n) s += L_s[n * LDL + tid] * rinv[n] * v[n];
      u[tid] = s;
    }
    if (tid == 0) {                                    // mahal part1, active_N
      float p1 = 0.f, an = 0.f;
      for (int n = 0; n < NOBS; ++n) {
        p1 += rinv[n] * v[n] * v[n];
        an += (rinv[n] != 0.f) ? 1.f : 0.f;
      }
      red[1] = p1; red[2] = an;
    }
    __syncthreads();

    chol64_wave0(Pp, &red[4]);                         // Pp -> Cp, logdet P_pred
    __syncthreads();
    trinv64_wave0(Pp, Ci);                             // Ci = Cp^{-1}
    __syncthreads();
    wmma_gemm64(T1, LDA, KDIM,                         // T1 = P_pred^{-1} = Ci^T Ci
        [&](int m, int p) { return Ci[p * LDA + m]; },
        [&](int p, int n) { return Ci[p * LDA + n]; });
    __syncthreads();
    for (int i = tid; i < KDIM * KDIM; i += NTHREADS) {// S = P^{-1} + G (into G)
      const int r = i >> 6, c = i & 63;
      G[r * LDA + c] += T1[r * LDA + c];
    }
    __syncthreads();
    chol64_wave0(G, &red[5]);                          // G -> Cs, logdet S
    __syncthreads();
    trinv64_wave0(G, Ci);                              // Ci = Cs^{-1}
    __syncthreads();
    wmma_gemm64(P_s, LDA, KDIM,                        // P_upd = S^{-1} = Ci^T Ci
        [&](int m, int p) { return Ci[p * LDA + m]; },
        [&](int p, int n) { return Ci[p * LDA + n]; });
    if (tid < KDIM) {                                  // w1 = Ci u (lower-tri)
      float s = 0.f;
      for (int p = 0; p <= tid; ++p) s += Ci[tid * LDA + p] * u[p];
      w1[tid] = s;
    }
    __syncthreads();
    if (tid < KDIM) {                                  // wv = Ci^T w1 = S^{-1} u
      float s = 0.f;
      for (int i = tid; i < KDIM; ++i) s += Ci[i * LDA + tid] * w1[i];
      wv[tid] = s;
    }
    __syncthreads();
    if (tid < KDIM) {                                  // state update + output
      const float zn = zp[tid] + wv[tid];
      z[tid] = zn;
      out[(size_t)t * (KDIM + 1) + tid] = zn;
    }
    if (tid == 0) {                                    // log-likelihood
      float uw = 0.f;
      for (int k = 0; k < KDIM; ++k) uw += u[k] * wv[k];
      const float mahal   = red[1] - uw;
      const float logdetF = red[3] + red[4] + red[5];
      float ll = -0.5f * (red[2] * LOG2PI + logdetF + mahal);
      if (!(ll == ll)) ll = -1e6f;                     // nan_to_num
      out[(size_t)t * (KDIM + 1) + KDIM] = ll;
    }
    __syncthreads();
  }
}

extern "C" void kernel_launch(void* const* d_in, const int* in_sizes, int n_in,
                              void* d_out, int out_size, void* d_ws, size_t ws_size,
                              hipStream_t stream) {
  const float* obs   = (const float*)d_in[0];   // [2048,256]
  const float* Lam   = (const float*)d_in[1];   // [2048,256,64]
  const float* A_raw = (const float*)d_in[2];   // [64,64]
  const float* lQ    = (const float*)d_in[3];   // [64]
  const float* lR    = (const float*)d_in[4];   // [256]
  float* out = (float*)d_out;                   // [2048,65]
  (void)in_sizes; (void)n_in; (void)out_size; (void)d_ws; (void)ws_size;

  const size_t smem =
      (size_t)(6 * KDIM * LDA + 2 * NOBS * LDL + 3 * NOBS + 8 * KDIM + 32) * sizeof(float);
  kalman_persistent<<<1, NTHREADS, smem, stream>>>(obs, Lam, A_raw, lQ, lR, out);
}